// InfraGNN_41455024341259
// MI455X (gfx1250) — compile-verified
//
#include <hip/hip_runtime.h>
#include <hip/hip_bf16.h>

typedef float v2f __attribute__((ext_vector_type(2)));
typedef float v8f __attribute__((ext_vector_type(8)));

#define NEG_SLOPE 0.2f

__device__ __forceinline__ float elu_f(float v) { return v > 0.f ? v : (__expf(v) - 1.f); }
__device__ __forceinline__ unsigned fkey(float f) {
    unsigned u = __float_as_uint(f);
    return (u & 0x80000000u) ? ~u : (u | 0x80000000u);
}
__device__ __forceinline__ float funkey(unsigned k) {
    unsigned u = (k & 0x80000000u) ? (k & 0x7FFFFFFFu) : ~k;
    return __uint_as_float(u);
}

// ---------------- generic zero fill ----------------
__global__ void k_zero(float* p, size_t n) {
    size_t i = (size_t)blockIdx.x * blockDim.x + threadIdx.x;
    if (i < n) p[i] = 0.f;
}

// ---------------- layer 1 GEMM (K=9) + fused attention dots ----------------
// one block (256 thr) per node; thread t computes h1[n, t] (t = head*64 + ch)
__global__ void k_gemm1(const float* __restrict__ x, const float* __restrict__ W1,
                        const float* __restrict__ as1, const float* __restrict__ ad1,
                        float* __restrict__ h1, float* __restrict__ a_src1,
                        float* __restrict__ a_dst1) {
    __shared__ float xs[9];
    __shared__ float ss[256], sd[256];
    int nid = blockIdx.x;
    int t = threadIdx.x;
    if (t < 9) xs[t] = x[(size_t)nid * 9 + t];
    __syncthreads();
    float acc = 0.f;
#pragma unroll
    for (int k = 0; k < 9; ++k) acc += xs[k] * W1[k * 256 + t];
    h1[(size_t)nid * 256 + t] = acc;
    ss[t] = acc * as1[t];
    sd[t] = acc * ad1[t];
    for (int off = 32; off >= 1; off >>= 1) {
        __syncthreads();
        if ((t & 63) < off) { ss[t] += ss[t + off]; sd[t] += sd[t + off]; }
    }
    __syncthreads();
    if ((t & 63) == 0) {
        a_src1[nid * 4 + (t >> 6)] = ss[t];
        a_dst1[nid * 4 + (t >> 6)] = sd[t];
    }
}

// ---------------- layer 1 edge passes (per edge-head, H=4) ----------------
__global__ void k_edge_max1(const int* __restrict__ ei, int E, int n,
                            const float* __restrict__ a_src1, const float* __restrict__ a_dst1,
                            float* __restrict__ e1, unsigned* __restrict__ m1) {
    int tid = blockIdx.x * blockDim.x + threadIdx.x;
    int tot = (E + n) * 4;
    if (tid >= tot) return;
    int i = tid >> 2, h = tid & 3;
    int s = (i < E) ? ei[i] : (i - E);
    int d = (i < E) ? ei[E + i] : (i - E);
    float e = a_src1[s * 4 + h] + a_dst1[d * 4 + h];
    e = e > 0.f ? e : NEG_SLOPE * e;
    e1[tid] = e;
    atomicMax(&m1[d * 4 + h], fkey(e));
}

__global__ void k_edge_sum1(const int* __restrict__ ei, int E, int n,
                            const unsigned* __restrict__ m1, float* __restrict__ e1,
                            float* __restrict__ denom1) {
    int tid = blockIdx.x * blockDim.x + threadIdx.x;
    int tot = (E + n) * 4;
    if (tid >= tot) return;
    int i = tid >> 2, h = tid & 3;
    int d = (i < E) ? ei[E + i] : (i - E);
    float w = __expf(e1[tid] - funkey(m1[d * 4 + h]));
    e1[tid] = w;
    atomicAdd(&denom1[d * 4 + h], w);
}

__global__ void k_edge_agg1(const int* __restrict__ ei, int E, int n,
                            const float* __restrict__ e1, const float* __restrict__ denom1,
                            const float* __restrict__ h1, float* __restrict__ agg1) {
    int tid = blockIdx.x * blockDim.x + threadIdx.x;
    int tot = (E + n) * 4;
    if (tid >= tot) return;
    int i = tid >> 2, h = tid & 3;
    int s = (i < E) ? ei[i] : (i - E);
    int d = (i < E) ? ei[E + i] : (i - E);
    float alpha = e1[tid] / (denom1[d * 4 + h] + 1e-16f);
    const float4* hs = (const float4*)(h1 + (size_t)s * 256 + h * 64);
    float* od = agg1 + (size_t)d * 256 + h * 64;
#pragma unroll
    for (int c = 0; c < 16; ++c) {
        float4 v = hs[c];
        atomicAdd(od + 4 * c + 0, v.x * alpha);
        atomicAdd(od + 4 * c + 1, v.y * alpha);
        atomicAdd(od + 4 * c + 2, v.z * alpha);
        atomicAdd(od + 4 * c + 3, v.w * alpha);
    }
}

__global__ void k_bias_elu1(float* __restrict__ agg1, const float* __restrict__ b1, size_t total) {
    size_t idx = (size_t)blockIdx.x * blockDim.x + threadIdx.x;
    if (idx >= total) return;
    float v = agg1[idx] + b1[idx & 255];
    agg1[idx] = elu_f(v);
}

// ---------------- layer 2 GEMM: [N,256] x [256,64] via V_WMMA_F32_16X16X4_F32 ----------------
// one wave computes a 64-row x 16-col output panel (4 row tiles) so the B fragment
// loaded per K-step is reused by 4 WMMAs: cuts W2 L2 traffic 4x vs tile-per-wave.
__global__ void k_gemm2_wmma(const float* __restrict__ A, const float* __restrict__ B,
                             float* __restrict__ C, int n) {
    int wave = (blockIdx.x * blockDim.x + threadIdx.x) >> 5;
    int lane = threadIdx.x & 31;
    int ntile = (n + 15) >> 4;        // 16-row tiles
    int ngrp = (ntile + 3) >> 2;      // groups of 4 row tiles
    int tg = wave >> 2;               // row group
    int tn = wave & 3;                // col tile (4 x 16 = 64 cols)
    if (tg >= ngrp) return;           // uniform per wave
    int half = lane >> 4;             // 0: lanes 0-15, 1: lanes 16-31
    int l = lane & 15;
    int r0 = tg * 64 + l;
    // clamp A rows for the remainder group (reads valid memory; stores guarded below)
    const float* a0 = A + (size_t)min(r0, n - 1) * 256;
    const float* a1 = A + (size_t)min(r0 + 16, n - 1) * 256;
    const float* a2 = A + (size_t)min(r0 + 32, n - 1) * 256;
    const float* a3 = A + (size_t)min(r0 + 48, n - 1) * 256;
    const float* bcol = B + tn * 16 + l;
    v8f acc0 = {}, acc1 = {}, acc2 = {}, acc3 = {};
    for (int k = 0; k < 256; k += 4) {
        int ko = k + 2 * half;  // ISA 7.12.2: half 0 -> K=k+0,k+1 ; half 1 -> K=k+2,k+3
        v2f b, x0, x1, x2, x3;
        b.x = bcol[(size_t)(ko + 0) * 64];
        b.y = bcol[(size_t)(ko + 1) * 64];
        x0.x = a0[ko]; x0.y = a0[ko + 1];
        x1.x = a1[ko]; x1.y = a1[ko + 1];
        x2.x = a2[ko]; x2.y = a2[ko + 1];
        x3.x = a3[ko]; x3.y = a3[ko + 1];
        acc0 = __builtin_amdgcn_wmma_f32_16x16x4_f32(false, x0, false, b, (short)0, acc0, false, false);
        acc1 = __builtin_amdgcn_wmma_f32_16x16x4_f32(false, x1, false, b, (short)0, acc1, false, false);
        acc2 = __builtin_amdgcn_wmma_f32_16x16x4_f32(false, x2, false, b, (short)0, acc2, false, false);
        acc3 = __builtin_amdgcn_wmma_f32_16x16x4_f32(false, x3, false, b, (short)0, acc3, false, false);
    }
    // C/D layout: VGPR r holds row M = r + 8*half, col N = l. n%16==0 so base<n => full tile.
    int col = tn * 16 + l;
#pragma unroll
    for (int t = 0; t < 4; ++t) {
        int base = tg * 64 + t * 16;
        if (base >= n) break;
        v8f acc = (t == 0) ? acc0 : (t == 1) ? acc1 : (t == 2) ? acc2 : acc3;
        float* crow = C + (size_t)base * 64 + col;
#pragma unroll
        for (int r = 0; r < 8; ++r) crow[(size_t)(r + 8 * half) * 64] = acc[r];
    }
}

// ---------------- layer 2 attention dots ----------------
__global__ void k_att2(const float* __restrict__ h2, const float* __restrict__ as2,
                       const float* __restrict__ ad2, float* __restrict__ a_src2,
                       float* __restrict__ a_dst2, int n) {
    int nid = blockIdx.x * blockDim.x + threadIdx.x;
    if (nid >= n) return;
    const float* row = h2 + (size_t)nid * 64;
    float s = 0.f, d = 0.f;
#pragma unroll 8
    for (int c = 0; c < 64; ++c) {
        float v = row[c];
        s += v * as2[c];
        d += v * ad2[c];
    }
    a_src2[nid] = s;
    a_dst2[nid] = d;
}

// ---------------- layer 2 edge passes (H=1) ----------------
__global__ void k_edge_max2(const int* __restrict__ ei, int E, int n,
                            const float* __restrict__ a_src, const float* __restrict__ a_dst,
                            float* __restrict__ e2, unsigned* __restrict__ m2) {
    int i = blockIdx.x * blockDim.x + threadIdx.x;
    int ET = E + n;
    if (i >= ET) return;
    int s = (i < E) ? ei[i] : (i - E);
    int d = (i < E) ? ei[E + i] : (i - E);
    float e = a_src[s] + a_dst[d];
    e = e > 0.f ? e : NEG_SLOPE * e;
    e2[i] = e;
    atomicMax(&m2[d], fkey(e));
}

__global__ void k_edge_sum2(const int* __restrict__ ei, int E, int n,
                            const unsigned* __restrict__ m2, float* __restrict__ e2,
                            float* __restrict__ denom2) {
    int i = blockIdx.x * blockDim.x + threadIdx.x;
    int ET = E + n;
    if (i >= ET) return;
    int d = (i < E) ? ei[E + i] : (i - E);
    float w = __expf(e2[i] - funkey(m2[d]));
    e2[i] = w;
    atomicAdd(&denom2[d], w);
}

__global__ void k_edge_agg2(const int* __restrict__ ei, int E, int n,
                            const float* __restrict__ e2, const float* __restrict__ denom2,
                            const float* __restrict__ h2, float* __restrict__ agg2) {
    int i = blockIdx.x * blockDim.x + threadIdx.x;
    int ET = E + n;
    if (i >= ET) return;
    int s = (i < E) ? ei[i] : (i - E);
    int d = (i < E) ? ei[E + i] : (i - E);
    float alpha = e2[i] / (denom2[d] + 1e-16f);
    const float4* hs = (const float4*)(h2 + (size_t)s * 64);
    float* od = agg2 + (size_t)d * 64;
#pragma unroll
    for (int c = 0; c < 16; ++c) {
        float4 v = hs[c];
        atomicAdd(od + 4 * c + 0, v.x * alpha);
        atomicAdd(od + 4 * c + 1, v.y * alpha);
        atomicAdd(od + 4 * c + 2, v.z * alpha);
        atomicAdd(od + 4 * c + 3, v.w * alpha);
    }
}

// ---------------- bias+ELU (layer 2) fused with classifier MLP + sigmoid ----------------
__global__ void k_final(const float* __restrict__ agg2, const float* __restrict__ b2,
                        const float* __restrict__ Wc1, const float* __restrict__ bc1,
                        const float* __restrict__ Wc2, const float* __restrict__ bc2,
                        float* __restrict__ out, int n) {
    __shared__ float sW[64 * 32];
    __shared__ float sb1[32], sW2[32];
    int t = threadIdx.x;
    for (int i = t; i < 64 * 32; i += 256) sW[i] = Wc1[i];
    if (t < 32) { sb1[t] = bc1[t]; sW2[t] = Wc2[t]; }
    __syncthreads();
    int nid = blockIdx.x * 256 + t;
    if (nid >= n) return;
    float h[64];
    const float* row = agg2 + (size_t)nid * 64;
#pragma unroll
    for (int c = 0; c < 64; ++c) h[c] = elu_f(row[c] + b2[c]);
    float logit = bc2[0];
    for (int j = 0; j < 32; ++j) {
        float z = sb1[j];
#pragma unroll
        for (int c = 0; c < 64; ++c) z += h[c] * sW[c * 32 + j];
        z = elu_f(z);
        logit += z * sW2[j];
    }
    out[nid] = 1.f / (1.f + __expf(-logit));
}

extern "C" void kernel_launch(void* const* d_in, const int* in_sizes, int n_in,
                              void* d_out, int out_size, void* d_ws, size_t ws_size,
                              hipStream_t stream) {
    const float* x   = (const float*)d_in[0];
    const int*   ei  = (const int*)d_in[1];
    const float* W1  = (const float*)d_in[2];
    const float* as1 = (const float*)d_in[3];
    const float* ad1 = (const float*)d_in[4];
    const float* b1  = (const float*)d_in[5];
    const float* W2  = (const float*)d_in[6];
    const float* as2 = (const float*)d_in[7];
    const float* ad2 = (const float*)d_in[8];
    const float* b2  = (const float*)d_in[9];
    const float* Wc1 = (const float*)d_in[10];
    const float* bc1 = (const float*)d_in[11];
    const float* Wc2 = (const float*)d_in[12];
    const float* bc2 = (const float*)d_in[13];
    float* out = (float*)d_out;

    const int N = in_sizes[0] / 9;
    const int E = in_sizes[1] / 2;
    const int ET = E + N;

    // ---- workspace layout (floats); layer-2 buffers overlay h1 (dead after agg1) ----
    float* ws = (float*)d_ws;
    size_t o = 0;
    float*    h1     = ws + o; o += (size_t)N * 256;
    float*    agg1   = ws + o; o += (size_t)N * 256;
    float*    e1     = ws + o; o += (size_t)ET * 4;
    float*    a_src1 = ws + o; o += (size_t)N * 4;
    float*    a_dst1 = ws + o; o += (size_t)N * 4;
    unsigned* m1     = (unsigned*)(ws + o); o += (size_t)N * 4;
    float*    denom1 = ws + o; o += (size_t)N * 4;
    // layer-2 region reuses h1's N*256 floats (needs N*128 + ET + 4N)
    float*    h2     = h1;
    float*    e2     = h1 + (size_t)N * 64;
    float*    a_src2 = e2 + ET;
    float*    a_dst2 = a_src2 + N;
    unsigned* m2     = (unsigned*)(a_dst2 + N);
    float*    denom2 = (float*)m2 + N;
    float*    agg2   = denom2 + N;

    const int B = 256;
    auto blocks = [](size_t total, int b) { return (unsigned)((total + b - 1) / b); };

    // ---- layer 1 ----
    size_t z1 = (size_t)N * 4 * 2;  // m1 (key 0 == -inf) + denom1 are adjacent
    k_zero<<<blocks(z1, B), B, 0, stream>>>((float*)m1, z1);
    k_zero<<<blocks((size_t)N * 256, B), B, 0, stream>>>(agg1, (size_t)N * 256);
    k_gemm1<<<N, 256, 0, stream>>>(x, W1, as1, ad1, h1, a_src1, a_dst1);
    k_edge_max1<<<blocks((size_t)ET * 4, B), B, 0, stream>>>(ei, E, N, a_src1, a_dst1, e1, m1);
    k_edge_sum1<<<blocks((size_t)ET * 4, B), B, 0, stream>>>(ei, E, N, m1, e1, denom1);
    k_edge_agg1<<<blocks((size_t)ET * 4, B), B, 0, stream>>>(ei, E, N, e1, denom1, h1, agg1);
    k_bias_elu1<<<blocks((size_t)N * 256, B), B, 0, stream>>>(agg1, b1, (size_t)N * 256);

    // ---- layer 2 (h1 now dead; overlay region live) ----
    int ntile = (N + 15) / 16;
    int ngrp = (ntile + 3) / 4;
    size_t waves = (size_t)ngrp * 4;          // 64-row x 16-col panels
    k_gemm2_wmma<<<blocks(waves * 32, B), B, 0, stream>>>(agg1, W2, h2, N);
    k_att2<<<blocks((size_t)N, B), B, 0, stream>>>(h2, as2, ad2, a_src2, a_dst2, N);
    k_zero<<<blocks((size_t)N * 2, B), B, 0, stream>>>((float*)m2, (size_t)N * 2); // m2+denom2
    k_zero<<<blocks((size_t)N * 64, B), B, 0, stream>>>(agg2, (size_t)N * 64);
    k_edge_max2<<<blocks((size_t)ET, B), B, 0, stream>>>(ei, E, N, a_src2, a_dst2, e2, m2);
    k_edge_sum2<<<blocks((size_t)ET, B), B, 0, stream>>>(ei, E, N, m2, e2, denom2);
    k_edge_agg2<<<blocks((size_t)ET, B), B, 0, stream>>>(ei, E, N, e2, denom2, h2, agg2);

    // ---- bias+ELU + classifier + sigmoid ----
    k_final<<<blocks((size_t)N, B), B, 0, stream>>>(agg2, b2, Wc1, bc1, Wc2, bc2, out, N);
}